// Attention_29678224015704
// MI455X (gfx1250) — compile-verified
//
#include <hip/hip_runtime.h>
#include <math.h>

// Sizes from the reference
#define LL 2048
#define BB 64
#define D_DEC 1024
#define D_ALIGN 512
#define D_ENC 1024

typedef __attribute__((ext_vector_type(2))) float v2f;
typedef __attribute__((ext_vector_type(8))) float v8f;

// ---------------------------------------------------------------------------
// Kernel 1: proj[b][a] = sum_k s_tm1[b][k] * sa_w[a][k] + sa_b[a]
// (64 x 512 output) via V_WMMA_F32_16X16X4_F32, one 16x16 tile per wave.
// 128 tiles total -> 32 blocks of 4 waves (128 threads).
// ---------------------------------------------------------------------------
__global__ void proj_wmma_kernel(const float* __restrict__ s_tm1,
                                 const float* __restrict__ sa_w,
                                 const float* __restrict__ sa_b,
                                 float* __restrict__ proj) {
  const int wave = threadIdx.x >> 5;
  const int lane = threadIdx.x & 31;
  const int tile = blockIdx.x * 4 + wave;   // 0..127
  const int tm   = tile >> 5;               // 0..3   (B / 16)
  const int tn   = tile & 31;               // 0..31  (D_ALIGN / 16)
  const int half = lane >> 4;               // 0 or 1 -> K pair {0,1} or {2,3}
  const int mr   = lane & 15;

  // A: 16x4 tile of s_tm1; lane holds row (tm*16+mr), K = k0 + 2*half + {0,1}
  const float* __restrict__ arow = s_tm1 + (size_t)(tm * 16 + mr) * D_DEC;
  // B: 4x16 tile; B[k][n] = sa_w[n][k]; lane holds column n = tn*16+mr
  const float* __restrict__ brow = sa_w + (size_t)(tn * 16 + mr) * D_DEC;

  v8f c = {};
  for (int k0 = 0; k0 < D_DEC; k0 += 4) {
    v2f a = *(const v2f*)(arow + k0 + 2 * half);
    v2f b = *(const v2f*)(brow + k0 + 2 * half);
    // 8 args: (neg_a, A, neg_b, B, c_mod, C, reuse_a, reuse_b)
    c = __builtin_amdgcn_wmma_f32_16x16x4_f32(false, a, false, b,
                                              (short)0, c, false, false);
  }

  // D layout: VGPR v, lane l -> M = v + 8*(l>=16), N = l&15
#pragma unroll
  for (int v = 0; v < 8; ++v) {
    const int M   = v + 8 * half;
    const int row = tm * 16 + M;     // batch index
    const int col = tn * 16 + mr;    // align-dim index
    proj[(size_t)row * D_ALIGN + col] = c[v] + sa_b[col];
  }
}

// ---------------------------------------------------------------------------
// Kernel 2: score[l][b] = sum_k tanh(proj[b][k] + uh[l][b][k]) * a1_w[k] + a1_b
// One wave per (l,b): streams 512 contiguous floats of uh, wave32 reduction.
// ---------------------------------------------------------------------------
__global__ void score_kernel(const float* __restrict__ proj,
                             const float* __restrict__ uh,
                             const float* __restrict__ a1_w,
                             const float* __restrict__ a1_b,
                             float* __restrict__ score) {
  const int wid  = (int)((blockIdx.x * blockDim.x + threadIdx.x) >> 5);
  const int lane = threadIdx.x & 31;
  const int l = wid >> 6;   // / BB
  const int b = wid & 63;   // % BB

  const float* __restrict__ u = uh + ((size_t)l * BB + b) * D_ALIGN;
  const float* __restrict__ p = proj + (size_t)b * D_ALIGN;

  float acc = 0.0f;
#pragma unroll
  for (int i = 0; i < D_ALIGN / 32; ++i) {
    const int k = i * 32 + lane;
    acc += tanhf(p[k] + u[k]) * a1_w[k];
  }
  // wave32 butterfly reduction
#pragma unroll
  for (int off = 16; off > 0; off >>= 1)
    acc += __shfl_xor(acc, off, 32);

  if (lane == 0)
    score[(size_t)l * BB + b] = acc + a1_b[0];
}

// ---------------------------------------------------------------------------
// Kernel 3: masked softmax over L for each column b (matches MaskSoftmax:
// max over ALL scores, exp(score-max)*mask, normalize by masked sum).
// One block (256 threads) per column.
// ---------------------------------------------------------------------------
__global__ void softmax_kernel(const float* __restrict__ score,
                               const float* __restrict__ xs_mask,
                               float* __restrict__ e_out) {
  const int b = blockIdx.x;
  const int t = threadIdx.x;
  __shared__ float red[256];

  float mx = -INFINITY;
#pragma unroll
  for (int i = 0; i < LL / 256; ++i) {
    const int l = t + i * 256;
    mx = fmaxf(mx, score[(size_t)l * BB + b]);
  }
  red[t] = mx;
  __syncthreads();
  for (int s = 128; s > 0; s >>= 1) {
    if (t < s) red[t] = fmaxf(red[t], red[t + s]);
    __syncthreads();
  }
  mx = red[0];
  __syncthreads();

  float vals[LL / 256];
  float sum = 0.0f;
#pragma unroll
  for (int i = 0; i < LL / 256; ++i) {
    const int l = t + i * 256;
    const float e = expf(score[(size_t)l * BB + b] - mx) *
                    xs_mask[(size_t)l * BB + b];
    vals[i] = e;
    sum += e;
  }
  red[t] = sum;
  __syncthreads();
  for (int s = 128; s > 0; s >>= 1) {
    if (t < s) red[t] += red[t + s];
    __syncthreads();
  }
  const float inv = 1.0f / red[0];
#pragma unroll
  for (int i = 0; i < LL / 256; ++i) {
    const int l = t + i * 256;
    e_out[(size_t)l * BB + b] = vals[i] * inv;
  }
}

// ---------------------------------------------------------------------------
// Kernel 4: attend[b][d] = sum_l e[l][b] * xs_h[l][b][d]
// One thread per (b,d); lanes cover consecutive d -> 128B coalesced per l.
// grid = BB * (D_ENC/256) blocks of 256 threads.
// ---------------------------------------------------------------------------
__global__ void attend_kernel(const float* __restrict__ e,
                              const float* __restrict__ xs_h,
                              float* __restrict__ attend) {
  const int b = blockIdx.x >> 2;                       // D_ENC/256 = 4
  const int d = (blockIdx.x & 3) * 256 + threadIdx.x;

  const float* __restrict__ xb = xs_h + (size_t)b * D_ENC + d;
  const float* __restrict__ eb = e + b;

  float acc = 0.0f;
#pragma unroll 4
  for (int l = 0; l < LL; ++l) {
    acc += eb[(size_t)l * BB] * xb[(size_t)l * BB * D_ENC];
  }
  attend[(size_t)b * D_ENC + d] = acc;
}

// ---------------------------------------------------------------------------
// Host launcher
// ---------------------------------------------------------------------------
extern "C" void kernel_launch(void* const* d_in, const int* in_sizes, int n_in,
                              void* d_out, int out_size, void* d_ws, size_t ws_size,
                              hipStream_t stream) {
  const float* s_tm1   = (const float*)d_in[0];
  const float* xs_h    = (const float*)d_in[1];
  const float* uh      = (const float*)d_in[2];
  const float* xs_mask = (const float*)d_in[3];
  const float* sa_w    = (const float*)d_in[4];
  const float* sa_b    = (const float*)d_in[5];
  const float* a1_w    = (const float*)d_in[6];
  const float* a1_b    = (const float*)d_in[7];

  float* proj  = (float*)d_ws;                                   // 64*512 f32 = 128 KB
  float* score = (float*)((char*)d_ws + (size_t)BB * D_ALIGN * 4); // 2048*64 f32 = 512 KB

  float* e_ij   = (float*)d_out;               // (L, B)
  float* attend = e_ij + (size_t)LL * BB;      // (B, D_ENC)

  // 1) proj GEMM via WMMA: 128 tiles, 4 waves/block
  proj_wmma_kernel<<<32, 128, 0, stream>>>(s_tm1, sa_w, sa_b, proj);

  // 2) scores: one wave per (l,b) -> L*B waves / 8 waves per block
  score_kernel<<<(LL * BB) / 8, 256, 0, stream>>>(proj, uh, a1_w, a1_b, score);

  // 3) masked softmax per column
  softmax_kernel<<<BB, 256, 0, stream>>>(score, xs_mask, e_ij);

  // 4) weighted sum of encoder states
  attend_kernel<<<BB * (D_ENC / 256), 256, 0, stream>>>(e_ij, xs_h, attend);
}